// CCA_SSG_26792005992870
// MI455X (gfx1250) — compile-verified
//
#include <hip/hip_runtime.h>
#include <cmath>

typedef float v2f __attribute__((ext_vector_type(2)));
typedef float v8f __attribute__((ext_vector_type(8)));

#define D 128
#define ALPHA 0.1f
#define EPS 1e-12f

// ---------------------------------------------------------------- fill (float4)
__global__ __launch_bounds__(256) void fill_kernel(float4* __restrict__ p, float v, int n4) {
    int i = blockIdx.x * blockDim.x + threadIdx.x;
    if (i < n4) p[i] = make_float4(v, v, v, v);
}

// ---------------------------------------------------------------- degree
__global__ __launch_bounds__(256) void degree_kernel(const int* __restrict__ dst,
                                                     float* __restrict__ deg, int ne) {
    int e = blockIdx.x * blockDim.x + threadIdx.x;
    if (e < ne) atomicAdd(&deg[dst[e]], 1.0f);
}

// deg -> norm in place: clip(deg,1)^-0.5
__global__ __launch_bounds__(256) void norm_kernel(float* __restrict__ norm, int n) {
    int i = blockIdx.x * blockDim.x + threadIdx.x;
    if (i < n) norm[i] = rsqrtf(fmaxf(norm[i], 1.0f));
}

// ---------------------------------------------------------------- edge scatter
// one wave (32 lanes) per edge; lane handles a float4 chunk of the 128-wide row.
// agg[dst] += feat[src] * norm[src]
__global__ __launch_bounds__(256) void scatter_kernel(const float* __restrict__ feat,
                                                      const float* __restrict__ norm,
                                                      const int* __restrict__ src,
                                                      const int* __restrict__ dst,
                                                      float* __restrict__ agg, int ne) {
    int t = blockIdx.x * blockDim.x + threadIdx.x;
    int e = t >> 5;
    if (e >= ne) return;
    int lane = t & 31;
    int s = src[e];
    int d = dst[e];
    float ns = norm[s];
    float4 x = *(const float4*)&feat[s * D + lane * 4];
    float* ap = &agg[d * D + lane * 4];
    atomicAdd(ap + 0, x.x * ns);
    atomicAdd(ap + 1, x.y * ns);
    atomicAdd(ap + 2, x.z * ns);
    atomicAdd(ap + 3, x.w * ns);
}

// ---------------------------------------------------------------- combine
// h = (1-ALPHA) * norm[row] * agg + ALPHA * feat0
__global__ __launch_bounds__(256) void combine_kernel(const float4* __restrict__ agg,
                                                      const float* __restrict__ norm,
                                                      const float4* __restrict__ feat0,
                                                      float4* __restrict__ h, int n4) {
    int i = blockIdx.x * blockDim.x + threadIdx.x;
    if (i >= n4) return;
    int row = i >> 5;  // 32 float4 per row of 128
    float s = (1.0f - ALPHA) * norm[row];
    float4 a = agg[i];
    float4 f = feat0[i];
    float4 r;
    r.x = s * a.x + ALPHA * f.x;
    r.y = s * a.y + ALPHA * f.y;
    r.z = s * a.z + ALPHA * f.z;
    r.w = s * a.w + ALPHA * f.w;
    h[i] = r;
}

// ---------------------------------------------------------------- WMMA GEMM + epilogue
// out = relu((1-beta)*H + beta*(H @ W) + bias), H: [nrows x 128], W: [128 x 128]
// block = 256 threads = 8 waves; wave w owns rows [blk*128 + w*16, +16), all 128 cols.
// W staged in LDS in K-pair-interleaved layout:
//   sW2[kp*128 + col] = { W[2*kp][col], W[2*kp+1][col] }   (kp = 0..63)
// so every B fragment is a single ds_load_b64 directly into the WMMA source pair.
__global__ __launch_bounds__(256) void gemm_kernel(const float* __restrict__ H,
                                                   const float* __restrict__ W,
                                                   const float* __restrict__ bias,
                                                   float* __restrict__ out,
                                                   int nrows, float beta) {
    __shared__ v2f sW2[(D / 2) * D];  // 64 KB
    int t = threadIdx.x;
    // cooperative staging: entry i = kp*128 + c  (coalesced scalar loads across threads)
    for (int i = t; i < (D / 2) * D; i += 256) {
        int kp = i >> 7;
        int c = i & (D - 1);
        v2f w;
        w.x = W[(2 * kp) * D + c];
        w.y = W[(2 * kp + 1) * D + c];
        sW2[i] = w;
    }
    __syncthreads();

    int wave = t >> 5;
    int lane = t & 31;
    int half = lane >> 4;   // 0: K pair {k,k+1}; 1: K pair {k+2,k+3}
    int lmod = lane & 15;
    int m0 = blockIdx.x * 128 + wave * 16;
    // clamp instead of predicating: always a legal load, epilogue guards stores
    int arow = min(m0 + lmod, nrows - 1);

    v8f acc[8] = {};
#pragma unroll 4
    for (int kk = 0; kk < D; kk += 4) {
        int kp = (kk >> 1) + half;                  // K-pair index for this lane
        v2f a = *(const v2f*)&H[arow * D + kk + 2 * half];
        const v2f* wb = &sW2[kp * D + lmod];        // fragments at +16 entries (128 B)
#pragma unroll
        for (int nt = 0; nt < 8; ++nt) {
            v2f b = wb[nt * 16];
            // D = A(16x4) * B(4x16) + C  -> v_wmma_f32_16x16x4_f32
            acc[nt] = __builtin_amdgcn_wmma_f32_16x16x4_f32(
                /*neg_a=*/false, a, /*neg_b=*/false, b,
                /*c_mod=*/(short)0, acc[nt], /*reuse_a=*/false, /*reuse_b=*/false);
        }
    }

    // epilogue: C/D layout -> VGPR j: M = j + 8*half, N = lane%16
    float omb = 1.0f - beta;
#pragma unroll
    for (int nt = 0; nt < 8; ++nt) {
        int c = nt * 16 + lmod;
        float bs = bias[c];
#pragma unroll
        for (int j = 0; j < 8; ++j) {
            int r = m0 + j + 8 * half;
            if (r < nrows) {
                float hv = H[r * D + c];
                float o = omb * hv + beta * acc[nt][j] + bs;
                out[r * D + c] = fmaxf(o, 0.0f);
            }
        }
    }
}

// ---------------------------------------------------------------- column stats (partials)
__global__ __launch_bounds__(256) void colstat_kernel(const float* __restrict__ x,
                                                      float* __restrict__ sums,
                                                      float* __restrict__ sumsq, int nrows) {
    int c = threadIdx.x & (D - 1);
    int streams_per_blk = blockDim.x / D;                 // 2
    int rstart = blockIdx.x * streams_per_blk + (threadIdx.x >> 7);
    int rstride = gridDim.x * streams_per_blk;
    float s = 0.0f, s2 = 0.0f;
    for (int r = rstart; r < nrows; r += rstride) {
        float v = x[r * D + c];
        s += v;
        s2 += v * v;
    }
    atomicAdd(&sums[c], s);
    atomicAdd(&sumsq[c], s2);
}

// sums/sumsq -> mean / rstd (unbiased, ddof=1), clipped at EPS
__global__ __launch_bounds__(128) void finalize_stats_kernel(const float* __restrict__ sums,
                                                             const float* __restrict__ sumsq,
                                                             float* __restrict__ mean,
                                                             float* __restrict__ rstd, int nrows) {
    int c = threadIdx.x;
    float n = (float)nrows;
    float m = sums[c] / n;
    float var = (sumsq[c] - n * m * m) / (n - 1.0f);
    float sd = sqrtf(fmaxf(var, 0.0f));
    mean[c] = m;
    rstd[c] = 1.0f / fmaxf(sd, EPS);
}

__global__ __launch_bounds__(256) void standardize_kernel(const float4* __restrict__ x,
                                                          const float* __restrict__ mean,
                                                          const float* __restrict__ rstd,
                                                          float4* __restrict__ out, int n4) {
    int i = blockIdx.x * blockDim.x + threadIdx.x;
    if (i >= n4) return;
    int c4 = (i & 31) * 4;
    float4 v = x[i];
    float4 m = *(const float4*)&mean[c4];
    float4 r = *(const float4*)&rstd[c4];
    float4 o;
    o.x = (v.x - m.x) * r.x;
    o.y = (v.y - m.y) * r.y;
    o.z = (v.z - m.z) * r.z;
    o.w = (v.w - m.w) * r.w;
    out[i] = o;
}

// ---------------------------------------------------------------- host orchestration
extern "C" void kernel_launch(void* const* d_in, const int* in_sizes, int n_in,
                              void* d_out, int out_size, void* d_ws, size_t ws_size,
                              hipStream_t stream) {
    const float* feat[2] = {(const float*)d_in[0], (const float*)d_in[1]};
    const int* src[2] = {(const int*)d_in[2], (const int*)d_in[4]};
    const int* dst[2] = {(const int*)d_in[3], (const int*)d_in[5]};
    const float* weights = (const float*)d_in[6];  // [L,128,128]
    const float* biases  = (const float*)d_in[7];  // [L,128]
    const int E[2] = {in_sizes[2], in_sizes[4]};

    const int N = in_sizes[0] / D;          // 100000
    const int L = in_sizes[7] / D;          // 4
    const int ND = N * D;
    const int ND4 = ND / 4;

    // workspace layout (floats)
    float* w = (float*)d_ws;
    float* normb = w;            w += N;        // degree -> norm
    float* agg   = w;            w += ND;
    float* hbuf  = w;            w += ND;
    float* fbuf  = w;            w += ND;
    float* sums  = w;            w += D;
    float* sumsq = w;            w += D;
    float* meanb = w;            w += D;
    float* rstdb = w;            w += D;

    const int TB = 256;
    const int gridND4 = (ND4 + TB - 1) / TB;

    for (int g = 0; g < 2; ++g) {
        int ne = E[g];
        // degree + symmetric norm
        fill_kernel<<<(N / 4 + TB - 1) / TB, TB, 0, stream>>>((float4*)normb, 0.0f, N / 4);
        degree_kernel<<<(ne + TB - 1) / TB, TB, 0, stream>>>(dst[g], normb, ne);
        norm_kernel<<<(N + TB - 1) / TB, TB, 0, stream>>>(normb, N);

        const float* cur = feat[g];
        for (int l = 0; l < L; ++l) {
            float beta = logf(1.0f / (float)(l + 1) + 1.0f);
            fill_kernel<<<gridND4, TB, 0, stream>>>((float4*)agg, 0.0f, ND4);
            scatter_kernel<<<(ne * 32 + TB - 1) / TB, TB, 0, stream>>>(cur, normb, src[g], dst[g], agg, ne);
            combine_kernel<<<gridND4, TB, 0, stream>>>((const float4*)agg, normb,
                                                       (const float4*)feat[g], (float4*)hbuf, ND4);
            gemm_kernel<<<(N + 127) / 128, TB, 0, stream>>>(hbuf, weights + l * D * D,
                                                            biases + l * D, fbuf, N, beta);
            cur = fbuf;
        }

        // standardize columns of fbuf -> d_out segment
        fill_kernel<<<1, 64, 0, stream>>>((float4*)sums, 0.0f, 64);  // zero sums+sumsq (2*128 floats)
        colstat_kernel<<<512, TB, 0, stream>>>(fbuf, sums, sumsq, N);
        finalize_stats_kernel<<<1, 128, 0, stream>>>(sums, sumsq, meanb, rstdb, N);
        standardize_kernel<<<gridND4, TB, 0, stream>>>((const float4*)fbuf, meanb, rstdb,
                                                       (float4*)((float*)d_out + (size_t)g * ND), ND4);
    }
}